// MyMultilevelDense_66159676228328
// MI455X (gfx1250) — compile-verified
//
#include <hip/hip_runtime.h>
#include <math.h>

#define B_N 2048
#define G_N 64
#define U_N 128
#define P_N 256
#define UP  (U_N * P_N)

#define CC      0.5413248546129181f   // log(expm1(1))
#define LOG2PI  1.8378770664093453f
#define LOG100  4.605170185988091f
#define LOG1E5  11.512925464970229f

typedef float v2f __attribute__((ext_vector_type(2)));
typedef float v8f __attribute__((ext_vector_type(8)));

__device__ __forceinline__ float bred256(float v, float* sm) {
  int t = threadIdx.x;
  sm[t] = v; __syncthreads();
  #pragma unroll
  for (int s = 128; s > 0; s >>= 1) {
    if (t < s) sm[t] += sm[t + s];
    __syncthreads();
  }
  float r = sm[0]; __syncthreads();
  return r;
}

// ---- K1: per-group sums over [G,U,P] -------------------------------------
// grpW[g*5+{0..4}] = { Sum sp(sig), Sum log(sp(sig)), Sum(sp^2+(mu-w0s)^2),
//                      Sum mu^2, Sum mu*mu0 }
__global__ void group_reduce_w(const float* __restrict__ wmk, const float* __restrict__ wsk,
                               const float* __restrict__ wmu0, const float* __restrict__ wsg0,
                               const float* __restrict__ epsw0, float* __restrict__ grpW) {
  __shared__ float sm[256];
  int g = blockIdx.x;
  const float* wm = wmk + (size_t)g * UP;
  const float* ws = wsk + (size_t)g * UP;
  float a0 = 0, a1 = 0, a2 = 0, a3 = 0, a4 = 0;
  for (int i = threadIdx.x; i < UP; i += 256) {
    float m = wm[i], s = ws[i];
    float mu0 = wmu0[i];
    float s1 = __expf(CC + s);
    float w0s = mu0 + __expf(CC + wsg0[i]) * epsw0[i];
    float d = m - w0s;
    a0 += s1;
    a1 += s;                 // log(sp(s)) = CC + s; add Dw*CC at the end
    a2 += s1 * s1 + d * d;
    a3 += m * m;
    a4 += m * mu0;
  }
  a0 = bred256(a0, sm); a1 = bred256(a1, sm); a2 = bred256(a2, sm);
  a3 = bred256(a3, sm); a4 = bred256(a4, sm);
  if (threadIdx.x == 0) {
    grpW[g * 5 + 0] = a0;
    grpW[g * 5 + 1] = (float)UP * CC + a1;
    grpW[g * 5 + 2] = a2;
    grpW[g * 5 + 3] = a3;
    grpW[g * 5 + 4] = a4;
  }
}

// ---- K2: per-group sums over [G,U] (bias) --------------------------------
__global__ void group_reduce_b(const float* __restrict__ bmk, const float* __restrict__ bsk,
                               const float* __restrict__ bmu0, const float* __restrict__ bsg0,
                               const float* __restrict__ epsb0, float* __restrict__ grpB) {
  __shared__ float sm[256];
  int g = blockIdx.x;
  float a0 = 0, a1 = 0, a2 = 0, a3 = 0, a4 = 0;
  for (int i = threadIdx.x; i < U_N; i += 256) {
    float m = bmk[g * U_N + i], s = bsk[g * U_N + i];
    float mu0 = bmu0[i];
    float s1 = __expf(CC + s);
    float b0s = mu0 + __expf(CC + bsg0[i]) * epsb0[i];
    float d = m - b0s;
    a0 += s1; a1 += s; a2 += s1 * s1 + d * d; a3 += m * m; a4 += m * mu0;
  }
  a0 = bred256(a0, sm); a1 = bred256(a1, sm); a2 = bred256(a2, sm);
  a3 = bred256(a3, sm); a4 = bred256(a4, sm);
  if (threadIdx.x == 0) {
    grpB[g * 5 + 0] = a0;
    grpB[g * 5 + 1] = (float)U_N * CC + a1;
    grpB[g * 5 + 2] = a2;
    grpB[g * 5 + 3] = a3;
    grpB[g * 5 + 4] = a4;
  }
}

// ---- K3: prior scalars ----------------------------------------------------
// scal = { Sum sp(wsig0), Sum wmu0^2, Sum w_z0_kl, Sum sp(bsig0), Sum bmu0^2, Sum b_z0_kl }
__global__ void prior_scalars(const float* __restrict__ wmu0, const float* __restrict__ wsg0,
                              const float* __restrict__ bmu0, const float* __restrict__ bsg0,
                              float* __restrict__ scal) {
  __shared__ float sm[256];
  float a0 = 0, a1 = 0, a2 = 0, b0 = 0, b1 = 0, b2 = 0;
  for (int i = threadIdx.x; i < UP; i += 256) {
    float s = wsg0[i], m = wmu0[i];
    float sp = __expf(CC + s);
    a0 += sp; a1 += m * m;
    a2 += LOG100 - (CC + s) + (sp * sp + m * m) * (1.0f / 20000.0f) - 0.5f;
  }
  for (int i = threadIdx.x; i < U_N; i += 256) {
    float s = bsg0[i], m = bmu0[i];
    float sp = __expf(CC + s);
    b0 += sp; b1 += m * m;
    b2 += LOG100 - (CC + s) + (sp * sp + m * m) * (1.0f / 20000.0f) - 0.5f;
  }
  a0 = bred256(a0, sm); a1 = bred256(a1, sm); a2 = bred256(a2, sm);
  b0 = bred256(b0, sm); b1 = bred256(b1, sm); b2 = bred256(b2, sm);
  if (threadIdx.x == 0) {
    scal[0] = a0; scal[1] = a1; scal[2] = a2;
    scal[3] = b0; scal[4] = b1; scal[5] = b2;
  }
}

// ---- K4: single-block: counts, group-sort perm+tiles, per-example scalars -
__global__ void __launch_bounds__(256)
batch_scalars(const int* __restrict__ gid, const float* __restrict__ gkw,
              const float* __restrict__ wtkmu, const float* __restrict__ wtksig,
              const float* __restrict__ btkmu, const float* __restrict__ btksig,
              const float* __restrict__ eps_wtau, const float* __restrict__ eps_btau,
              const float* __restrict__ grpW, const float* __restrict__ grpB,
              const float* __restrict__ scal,
              int* __restrict__ perm, int* __restrict__ tileG, int* __restrict__ tileStart,
              int* __restrict__ tileCnt, int* __restrict__ numTiles,
              float* __restrict__ out_scalars) {
  __shared__ float sm[256];
  __shared__ int cnt[G_N], cur[G_N];
  __shared__ float sW[G_N * 5], sB[G_N * 5];
  int t = threadIdx.x;
  if (t < G_N) cnt[t] = 0;
  __syncthreads();
  for (int b = t; b < B_N; b += 256) atomicAdd(&cnt[gid[b]], 1);
  __syncthreads();
  if (t == 0) {
    int o = 0, nt = 0;
    for (int g = 0; g < G_N; ++g) {
      cur[g] = o;
      for (int s = 0; s < cnt[g]; s += 16) {
        tileG[nt] = g; tileStart[nt] = o + s;
        int c = cnt[g] - s; tileCnt[nt] = (c > 16) ? 16 : c;
        ++nt;
      }
      o += cnt[g];
    }
    *numTiles = nt;
  }
  __syncthreads();
  for (int b = t; b < B_N; b += 256) {
    int p = atomicAdd(&cur[gid[b]], 1);
    perm[p] = b;
  }
  for (int i = t; i < G_N * 5; i += 256) { sW[i] = grpW[i]; sB[i] = grpB[i]; }
  __syncthreads();

  // batch-global scalars from the reference's einsum('bij,bij->') etc.
  float m2w = 0, crw = 0, m2b = 0, crb = 0, inv = 0;
  if (t < G_N) {
    float c = (float)cnt[t];
    m2w = c * sW[t * 5 + 3]; crw = c * sW[t * 5 + 4];
    m2b = c * sB[t * 5 + 3]; crb = c * sB[t * 5 + 4];
    inv = 1.0f / gkw[t];
  }
  m2w = bred256(m2w, sm); crw = bred256(crw, sm);
  m2b = bred256(m2b, sm); crb = bred256(crb, sm);
  inv = bred256(inv, sm);

  float Swsig0 = scal[0], Swmu0sq = scal[1], klwz0 = scal[2];
  float Sbsig0 = scal[3], Sbmu0sq = scal[4], klbz0 = scal[5];
  const float Dw = (float)UP, Db = (float)U_N;

  float aTau = 0, aWlp = 0, aBlp = 0, aK7 = 0, aK8 = 0;
  for (int b = t; b < B_N; b += 256) {
    int g = gid[b];
    float gk = gkw[g];
    float wts = __expf(CC + wtksig[g]);
    float bts = __expf(CC + btksig[g]);
    float wt = wtkmu[g] + wts * eps_wtau[b]; float wt2 = wt * wt;
    float bt = btkmu[g] + bts * eps_btau[b]; float bt2 = bt * bt;
    float wtkl = LOG100 - (CC + wtksig[g]) + (wts * wts + wtkmu[g] * wtkmu[g]) * (1.0f / 20000.0f) - 0.5f;
    float btkl = LOG1E5 - (CC + btksig[g]) + (bts * bts + btkmu[g] * btkmu[g]) * (1.0f / 2.0e10f) - 0.5f;
    aTau += gk * (wtkl + btkl);
    float lwt2 = __logf(wt2);
    float lbt2 = __logf(bt2);
    float Tw = sW[g * 5 + 0] + Swsig0 + m2w + Swmu0sq - 2.0f * crw;
    aWlp += gk * (-0.5f * Dw * LOG2PI - 0.5f * Dw * lwt2 - 0.5f / wt2 * Tw);
    float Tb = sB[g * 5 + 0] + Sbsig0 + m2b + Sbmu0sq - 2.0f * crb;
    aBlp += gk * (-0.5f * Db * LOG2PI - 0.5f * Db * lbt2 - 0.5f / bt2 * Tb);
    aK7 += gk * (Dw * lwt2 - sW[g * 5 + 1] + sW[g * 5 + 2] / (2.0f * wt2 * wt2) - 0.5f * Dw);
    aK8 += gk * (Db * lbt2 - sB[g * 5 + 1] + sB[g * 5 + 2] / (2.0f * bt2 * bt2) - 0.5f * Db);
  }
  aTau = bred256(aTau, sm); aWlp = bred256(aWlp, sm); aBlp = bred256(aBlp, sm);
  aK7 = bred256(aK7, sm); aK8 = bred256(aK8, sm);
  if (t == 0) {
    float pk = 1.0f / inv;
    out_scalars[0] = klwz0 * pk + klbz0 * pk + aTau - aWlp - aBlp;  // kl_losses
    out_scalars[1] = aK7;
    out_scalars[2] = aK8;
  }
}

// ---- K5: grouped GEMM (mean part) via V_WMMA_F32_16X16X4_F32 -------------
// out[b,u] = Sum_p w_mu_k[g,u,p] * x[b,p] for 16-example x 16-unit tiles.
__global__ void __launch_bounds__(32)
gemm_mu_wmma(const float* __restrict__ x, const float* __restrict__ wmk,
             const int* __restrict__ perm, const int* __restrict__ tileG,
             const int* __restrict__ tileStart, const int* __restrict__ tileCnt,
             const int* __restrict__ numTiles, float* __restrict__ out) {
  int tIdx = blockIdx.x;
  if (tIdx >= *numTiles) return;                 // uniform per block
  int g = tileG[tIdx], base = tileStart[tIdx], cnt = tileCnt[tIdx];
  int l = threadIdx.x;
  int half = l >> 4, m = l & 15;
  int bi = perm[base + ((m < cnt) ? m : 0)];     // pad rows duplicate row 0
  const float* xrow = x + (size_t)bi * P_N + 2 * half;     // A: M=m, K pairs by half
  int u = blockIdx.y * 16 + m;
  const float* wrow = wmk + ((size_t)g * U_N + u) * P_N + 2 * half;  // B: N=u
  v8f c = {0, 0, 0, 0, 0, 0, 0, 0};
  #pragma unroll 4
  for (int p0 = 0; p0 < P_N; p0 += 4) {
    float2 av = *(const float2*)(xrow + p0);
    float2 bv = *(const float2*)(wrow + p0);
    v2f a; a[0] = av.x; a[1] = av.y;
    v2f b; b[0] = bv.x; b[1] = bv.y;
    c = __builtin_amdgcn_wmma_f32_16x16x4_f32(false, a, false, b, (short)0, c, false, false);
  }
  #pragma unroll
  for (int r = 0; r < 8; ++r) {
    int mr = r + half * 8;                       // D layout: VGPR r -> M = r (+8)
    if (mr < cnt) {
      int bo = perm[base + mr];
      out[(size_t)bo * U_N + u] = c[r];
    }
  }
}

// ---- K6: HBM-bound stream of eps_w: noise part + bias, fused onto out ----
__global__ void __launch_bounds__(256)
stream_out(const float* __restrict__ x, const int* __restrict__ gid,
           const float* __restrict__ wsk, const float* __restrict__ bmk,
           const float* __restrict__ bsk, const float* __restrict__ eps_w,
           const float* __restrict__ eps_b, float* __restrict__ out) {
  __shared__ float4 xs4[P_N / 4];
  int b = blockIdx.x;
  int tid = threadIdx.x;
  ((float*)xs4)[tid] = x[(size_t)b * P_N + tid];
  __syncthreads();
  int g = gid[b];
  int wave = tid >> 5, lane = tid & 31;
  for (int uu = 0; uu < 16; ++uu) {
    int u = wave * 16 + uu;
    const float* erow = eps_w + ((size_t)b * U_N + u) * P_N;
    const float* srow = wsk + ((size_t)g * U_N + u) * P_N;
    float acc = 0.0f;
    #pragma unroll
    for (int j = 0; j < 2; ++j) {
      int p0 = j * 128 + lane * 4;
      float4 e = *(const float4*)(erow + p0);
      float4 s = *(const float4*)(srow + p0);
      float4 xv = xs4[p0 >> 2];
      acc += e.x * __expf(CC + s.x) * xv.x;
      acc += e.y * __expf(CC + s.y) * xv.y;
      acc += e.z * __expf(CC + s.z) * xv.z;
      acc += e.w * __expf(CC + s.w) * xv.w;
    }
    #pragma unroll
    for (int off = 16; off > 0; off >>= 1)
      acc += __shfl_xor(acc, off, 32);
    if (lane == 0) {
      size_t o = (size_t)b * U_N + u;
      float bmg = bmk[g * U_N + u];
      float bsg = __expf(CC + bsk[g * U_N + u]);
      out[o] = out[o] + acc + bmg + bsg * eps_b[o];   // add onto WMMA mean part
    }
  }
}

extern "C" void kernel_launch(void* const* d_in, const int* in_sizes, int n_in,
                              void* d_out, int out_size, void* d_ws, size_t ws_size,
                              hipStream_t stream) {
  const float* x           = (const float*)d_in[0];
  const int*   gid         = (const int*)d_in[1];
  const float* w_mu_k      = (const float*)d_in[2];
  const float* w_sigma_k   = (const float*)d_in[3];
  const float* w_mu0       = (const float*)d_in[4];
  const float* w_sigma0    = (const float*)d_in[5];
  const float* w_tau_k_mu  = (const float*)d_in[6];
  const float* w_tau_k_sig = (const float*)d_in[7];
  const float* b_mu_k      = (const float*)d_in[8];
  const float* b_sigma_k   = (const float*)d_in[9];
  const float* b_mu0       = (const float*)d_in[10];
  const float* b_sigma0    = (const float*)d_in[11];
  const float* b_tau_k_mu  = (const float*)d_in[12];
  const float* b_tau_k_sig = (const float*)d_in[13];
  const float* gkw         = (const float*)d_in[14];
  const float* eps_w       = (const float*)d_in[15];
  const float* eps_b       = (const float*)d_in[16];
  const float* eps_w0      = (const float*)d_in[17];
  const float* eps_b0      = (const float*)d_in[18];
  const float* eps_wtau    = (const float*)d_in[19];
  const float* eps_btau    = (const float*)d_in[20];
  float* out = (float*)d_out;

  // scratch layout (~20 KB total)
  int*   iws = (int*)d_ws;
  float* fws = (float*)d_ws;
  int* perm = iws;            // [2048]
  int* tG   = iws + 2048;     // [256]
  int* tS   = iws + 2304;     // [256]
  int* tC   = iws + 2560;     // [256]
  int* nT   = iws + 2816;     // [1]
  float* grpW = fws + 4096;   // [64*5]
  float* grpB = fws + 4416;   // [64*5]
  float* scal = fws + 4736;   // [6]

  group_reduce_w<<<G_N, 256, 0, stream>>>(w_mu_k, w_sigma_k, w_mu0, w_sigma0, eps_w0, grpW);
  group_reduce_b<<<G_N, 256, 0, stream>>>(b_mu_k, b_sigma_k, b_mu0, b_sigma0, eps_b0, grpB);
  prior_scalars<<<1, 256, 0, stream>>>(w_mu0, w_sigma0, b_mu0, b_sigma0, scal);
  batch_scalars<<<1, 256, 0, stream>>>(gid, gkw, w_tau_k_mu, w_tau_k_sig, b_tau_k_mu, b_tau_k_sig,
                                       eps_wtau, eps_btau, grpW, grpB, scal,
                                       perm, tG, tS, tC, nT, out + (size_t)B_N * U_N);
  dim3 gg(192, U_N / 16);  // max tiles = sum ceil(cnt_g/16) <= 188
  gemm_mu_wmma<<<gg, 32, 0, stream>>>(x, w_mu_k, perm, tG, tS, tC, nT, out);
  stream_out<<<B_N, 256, 0, stream>>>(x, gid, w_sigma_k, b_mu_k, b_sigma_k, eps_w, eps_b, out);
}